// ScaledDotProductAttention_29892972380335
// MI455X (gfx1250) — compile-verified
//
#include <hip/hip_runtime.h>

// Problem constants (match reference setup_inputs): B=16, S=2048, D=64, fp32.
#define B_ 16
#define S_ 2048
#define D_ 64
#define NEG_ (-1000000000.0f)
#define SCALE_ 0.125f          // 1/sqrt(64)

#define PS_STRIDE 2052         // 2048 + 4 pad floats: breaks 2048 % 64 == 0 bank clash
#define TKC_ 512               // V rows staged per TDM chunk
#define VS_STRIDE 68           // 64 + 4 pad dwords, produced by TDM pad feature

typedef __attribute__((ext_vector_type(2))) float v2f;
typedef __attribute__((ext_vector_type(8))) float v8f;
typedef __attribute__((ext_vector_type(4))) unsigned int v4u;
typedef __attribute__((ext_vector_type(8))) int v8i;
typedef __attribute__((ext_vector_type(4))) int v4i;

// ---------------------------------------------------------------------------
// Fully fused attention: one workgroup (16 waves, 512 threads) owns one
// 16-row strip of one batch. Scores -> LDS, softmax in LDS (attn written to
// HBM exactly once), then P @ V with TDM-staged V chunks and split-K waves.
// Grid: 16 batches * 128 strips = 2048 workgroups.
// ---------------------------------------------------------------------------
__global__ __launch_bounds__(512) void fused_attention_kernel(
    const float* __restrict__ q, const float* __restrict__ k,
    const float* __restrict__ v, const int* __restrict__ mask,
    float* __restrict__ attn, float* __restrict__ out)
{
    __shared__ float Ps[16 * PS_STRIDE];      // 131,328 B: strip of scores / P
    __shared__ float Vs[TKC_ * VS_STRIDE];    // 139,264 B: padded V chunk
    __shared__ float red[16 * D_];            //   4,096 B: split-K reduction tile

    const int tid  = threadIdx.x;
    const int wv   = tid >> 5;                // 16 waves
    const int lane = tid & 31;
    const int row  = lane & 15;
    const int half = lane >> 4;               // 0: K-offsets {0,1}, 1: {2,3}

    const int b  = blockIdx.x >> 7;           // 128 strips per batch
    const int mt = blockIdx.x & 127;
    const int s0 = mt * 16;

    const float* __restrict__ Q  = q + (size_t)b * S_ * D_;
    const float* __restrict__ K  = k + (size_t)b * S_ * D_;
    const float* __restrict__ Vb = v + (size_t)b * S_ * D_;
    const int*   __restrict__ M  = mask + (size_t)b * S_ * S_;
    float* __restrict__ Arow     = attn + (size_t)b * S_ * S_;
    float* __restrict__ O        = out  + (size_t)b * S_ * D_;

    // ======================= Phase 1: masked scores =======================
    // Wave wv owns columns [wv*128, wv*128+128), processed as two passes of
    // 64 columns (4 accumulators each) to stay within the VGPR budget.
    for (int nh = 0; nh < 2; ++nh) {
        const int n0 = wv * 128 + nh * 64;
        v8f accS[4] = {};
#pragma unroll
        for (int kk = 0; kk < D_; kk += 4) {
            // A tile 16x4 of Q: lane holds row (s0+row), K = kk + 2*half + {0,1}
            v2f a = *(const v2f*)(Q + (size_t)(s0 + row) * D_ + kk + 2 * half);
#pragma unroll
            for (int j = 0; j < 4; ++j) {
                // B tile = K^T chunk: lane holds column (n0 + j*16 + row)
                v2f bb = *(const v2f*)(K + (size_t)(n0 + j * 16 + row) * D_
                                         + kk + 2 * half);
                accS[j] = __builtin_amdgcn_wmma_f32_16x16x4_f32(
                    false, a, false, bb, (short)0, accS[j], false, false);
            }
        }
#pragma unroll
        for (int j = 0; j < 4; ++j) {
#pragma unroll
            for (int i = 0; i < 8; ++i) {
                const int mloc = i + 8 * half;          // strip-local row
                const int n    = n0 + j * 16 + row;
                const float maskAdd = M[(size_t)(s0 + mloc) * S_ + n] ? 0.0f : NEG_;
                Ps[mloc * PS_STRIDE + n] = accS[j][i] * SCALE_ + maskAdd;
            }
        }
    }
    __syncthreads();

    // ===================== Phase 2: softmax (wave wv = row wv) ============
    {
        float vals[64];
        float mx = -INFINITY;
        float* __restrict__ pr = &Ps[wv * PS_STRIDE];
#pragma unroll
        for (int i = 0; i < 16; ++i) {
            float4 vq = *(const float4*)&pr[i * 128 + lane * 4];
            vals[4*i+0] = vq.x; vals[4*i+1] = vq.y;
            vals[4*i+2] = vq.z; vals[4*i+3] = vq.w;
            mx = fmaxf(mx, fmaxf(fmaxf(vq.x, vq.y), fmaxf(vq.z, vq.w)));
        }
#pragma unroll
        for (int o = 16; o > 0; o >>= 1) mx = fmaxf(mx, __shfl_xor(mx, o, 32));
        float s = 0.0f;
#pragma unroll
        for (int i = 0; i < 64; ++i) { vals[i] = __expf(vals[i] - mx); s += vals[i]; }
#pragma unroll
        for (int o = 16; o > 0; o >>= 1) s += __shfl_xor(s, o, 32);
        const float inv = 1.0f / s;
        float* __restrict__ ag = &Arow[(size_t)(s0 + wv) * S_];
#pragma unroll
        for (int i = 0; i < 16; ++i) {
            float4 vq;
            vq.x = vals[4*i+0] * inv; vq.y = vals[4*i+1] * inv;
            vq.z = vals[4*i+2] * inv; vq.w = vals[4*i+3] * inv;
            *(float4*)&pr[i * 128 + lane * 4] = vq;      // normalized P for phase 3
            *(float4*)&ag[i * 128 + lane * 4] = vq;      // attn to HBM, exactly once
        }
    }
    __syncthreads();

    // ======================= Phase 3: out = P @ V =========================
    for (int i = tid; i < 16 * D_; i += 512) red[i] = 0.0f;

    v8f accO[4] = {};
    for (int c = 0; c < S_ / TKC_; ++c) {
        const int tbase = c * TKC_;
        __syncthreads();   // Vs safe to overwrite; red zero visible later

#if __has_builtin(__builtin_amdgcn_tensor_load_to_lds)
        if (tid == 0) {
            // ---- Tensor DMA Descriptor (CDNA5 ISA) ----
            const unsigned long long gaddr =
                (unsigned long long)(const void*)(Vb + (size_t)tbase * D_);
            const unsigned int ldsAddr = (unsigned int)(unsigned long long)(&Vs[0]);
            // Group 0: count=1 | lds_addr | global_addr[56:0] | type=2
            v4u g0 = { 1u,
                       ldsAddr,
                       (unsigned int)(gaddr & 0xFFFFFFFFull),
                       (unsigned int)((gaddr >> 32) & 0x1FFFFFFull) | (2u << 30) };
            // Group 1: data_size=2 (4B); pad_enable, pad_interval=5 (64 dwords),
            // pad_amount=3 (4 dwords) -> LDS row stride 68 dwords.
            // tensor 64 x 512, tile 64 x 512, stride0 = 64.
            v8i g1 = { (int)((2u << 16) | (1u << 20) | (5u << 22) | (3u << 25)),
                       (int)((unsigned)D_ << 16),    // [63:48]  tensor_dim0 lo16
                       (int)((unsigned)TKC_ << 16),  // [95:80]  tensor_dim1 lo16
                       (int)((unsigned)D_ << 16),    // [127:112] tile_dim0
                       (int)(unsigned)TKC_,          // [143:128] tile_dim1
                       (int)(unsigned)D_,            // [191:160] tensor_dim0_stride lo32
                       0, 0 };
            v4i g2 = { 0, 0, 0, 0 };
            v4i g3 = { 0, 0, 0, 0 };
            v8i g4 = { 0, 0, 0, 0, 0, 0, 0, 0 };
            __builtin_amdgcn_tensor_load_to_lds(g0, g1, g2, g3, g4, 0);
            __builtin_amdgcn_s_wait_tensorcnt(0);
        }
#else
        for (int i = tid; i < TKC_ * D_; i += 512)
            Vs[(i / D_) * VS_STRIDE + (i % D_)] = Vb[(size_t)tbase * D_ + i];
#endif
        __syncthreads();

        // Split-K: wave wv handles rows [wv*32, wv*32+32) of this chunk.
#pragma unroll
        for (int kk = 0; kk < 32; kk += 4) {
            const int kloc = wv * 32 + kk;
            // A tile 16x4 of P from LDS: row `row`, K = tbase+kloc+2*half+{0,1}
            v2f a = *(const v2f*)&Ps[row * PS_STRIDE + tbase + kloc + 2 * half];
            const int kr = kloc + 2 * half;
#pragma unroll
            for (int j = 0; j < 4; ++j) {
                v2f bb;
                bb.x = Vs[kr * VS_STRIDE + j * 16 + row];
                bb.y = Vs[(kr + 1) * VS_STRIDE + j * 16 + row];
                accO[j] = __builtin_amdgcn_wmma_f32_16x16x4_f32(
                    false, a, false, bb, (short)0, accO[j], false, false);
            }
        }
    }

    // Cross-wave split-K reduction via LDS float atomics.
#pragma unroll
    for (int j = 0; j < 4; ++j) {
#pragma unroll
        for (int i = 0; i < 8; ++i) {
            atomicAdd(&red[(i + 8 * half) * D_ + j * 16 + row], accO[j][i]);
        }
    }
    __syncthreads();

    for (int i = tid; i < 16 * D_; i += 512) {
        O[(size_t)(s0 + (i >> 6)) * D_ + (i & 63)] = red[i];
    }
}

// ---------------------------------------------------------------------------
extern "C" void kernel_launch(void* const* d_in, const int* in_sizes, int n_in,
                              void* d_out, int out_size, void* d_ws, size_t ws_size,
                              hipStream_t stream) {
    const float* q    = (const float*)d_in[0];
    const float* k    = (const float*)d_in[1];
    const float* v    = (const float*)d_in[2];
    const int*   mask = (const int*)d_in[3];

    float* out  = (float*)d_out;                       // [B,S,D] first
    float* attn = out + (size_t)B_ * S_ * D_;          // then [B,S,S]

    // One workgroup per (batch, 16-row strip): 16 * 128 = 2048 blocks.
    fused_attention_kernel<<<2048, 512, 0, stream>>>(q, k, v, mask, attn, out);
}